// PrimaryCaps_69320772157874
// MI455X (gfx1250) — compile-verified
//
#include <hip/hip_runtime.h>

typedef __attribute__((ext_vector_type(2))) float v2f;
typedef __attribute__((ext_vector_type(4))) float v4f;
typedef __attribute__((ext_vector_type(8))) float v8f;

#define NCAPS   4096
#define INDIM   512
#define CAPSLEN 16
#define BATCH   64
// 512 + 4 pad: bank = (n*516 + k) % 64; lanes cover 4n+koff{+0,+1} -> all 64
// banks hit exactly once per b64 wave read. Conflict-free.
#define LDS_STRIDE 516

__global__ __launch_bounds__(128)
void primarycaps_wmma_f32_kernel(const float* __restrict__ x,
                                 const float* __restrict__ W,
                                 const float* __restrict__ bias,
                                 float* __restrict__ out)
{
    __shared__ float ldsW[CAPSLEN * LDS_STRIDE];   // W_c transposed: [n][k]

    const int c   = blockIdx.x;        // capsule id
    const int tid = threadIdx.x;

    // ---- Stage W_c (512 x 16, row-major over [k][n]) transposed into LDS.
    // 8192 floats / 128 threads = 64 each, done as 16 coalesced b128 loads.
    const float* Wc = W + (size_t)c * (INDIM * CAPSLEN);
    #pragma unroll
    for (int i = 0; i < 16; ++i) {
        int e = 4 * tid + 512 * i;               // flat element = k*16 + n
        v4f w4 = *reinterpret_cast<const v4f*>(Wc + e);
        int k = e >> 4;
        int n = e & 15;
        ldsW[(n + 0) * LDS_STRIDE + k] = w4.x;
        ldsW[(n + 1) * LDS_STRIDE + k] = w4.y;
        ldsW[(n + 2) * LDS_STRIDE + k] = w4.z;
        ldsW[(n + 3) * LDS_STRIDE + k] = w4.w;
    }
    __syncthreads();

    // ---- Per-wave 16x16 output tile, K accumulated 4 at a time via WMMA.
    const int wave = tid >> 5;          // m-tile: batch rows [wave*16, wave*16+16)
    const int lane = tid & 31;
    const int half = lane >> 4;         // 0 -> K{0,1}, 1 -> K{2,3} (A & B layout)
    const int l16  = lane & 15;         // A: row M ; B: col N ; D: col N
    const int koff = half * 2;

    // A fragment source: x[brow, c, k..k+1], brow = wave*16 + l16 (both halves
    // read the same 16 rows at different k offsets) -> one global_load_b64.
    const float* xrow = x + ((size_t)(wave * 16 + l16) * NCAPS + c) * INDIM + koff;
    // B fragment source: Wt[n][k..k+1] -> one ds_load_b64.
    const float* brow = &ldsW[l16 * LDS_STRIDE + koff];

    v8f acc = {};
    #pragma unroll 8
    for (int k = 0; k < INDIM; k += 4) {
        v2f a = *reinterpret_cast<const v2f*>(xrow + k);
        v2f b = *reinterpret_cast<const v2f*>(brow + k);
        // D = A(16x4 f32) x B(4x16 f32) + C(16x16 f32)
        acc = __builtin_amdgcn_wmma_f32_16x16x4_f32(
            /*neg_a=*/false, a, /*neg_b=*/false, b,
            /*c_mod=*/(short)0, acc, /*reuse_a=*/false, /*reuse_b=*/false);
    }

    // ---- Epilogue: bias + store. D layout: VGPR v -> M = v + half*8, N = l16.
    const float bv = bias[c * CAPSLEN + l16];
    #pragma unroll
    for (int v = 0; v < 8; ++v) {
        int m = wave * 16 + half * 8 + v;                      // batch row
        out[((size_t)m * NCAPS + c) * CAPSLEN + l16] = acc[v] + bv;
    }
}

extern "C" void kernel_launch(void* const* d_in, const int* in_sizes, int n_in,
                              void* d_out, int out_size, void* d_ws, size_t ws_size,
                              hipStream_t stream) {
    const float* x    = (const float*)d_in[0];  // [64, 4096, 512]
    const float* W    = (const float*)d_in[1];  // [4096, 512, 16]
    const float* bias = (const float*)d_in[2];  // [4096, 16]
    float* out = (float*)d_out;                 // [64, 4096, 16]

    primarycaps_wmma_f32_kernel<<<NCAPS, 128, 0, stream>>>(x, W, bias, out);
}